// GraphSAGELayer_7284264534191
// MI455X (gfx1250) — compile-verified
//
#include <hip/hip_runtime.h>
#include <math.h>

typedef float v2f __attribute__((ext_vector_type(2)));
typedef float v8f __attribute__((ext_vector_type(8)));

#define N_NODES 50000
#define D_IN 256
#define D_OUT 256
#define EPS_F 1e-12f

// ---------------------------------------------------------------------------
// Kernel 1: zero the workspace (neigh_sum + deg)
// ---------------------------------------------------------------------------
__global__ void sage_zero_kernel(float* __restrict__ ws, long n) {
    long i = (long)blockIdx.x * blockDim.x + threadIdx.x;
    long stride = (long)gridDim.x * blockDim.x;
    for (; i < n; i += stride) ws[i] = 0.0f;
}

// ---------------------------------------------------------------------------
// Kernel 2: scatter-add neighbor features. One wave32 per edge:
// 8 iterations x 32 lanes cover the 256-wide feature row (coalesced 128B).
// ---------------------------------------------------------------------------
__global__ void sage_scatter_kernel(const float* __restrict__ x,
                                    const int* __restrict__ src,
                                    const int* __restrict__ dst,
                                    float* __restrict__ nsum,
                                    float* __restrict__ deg,
                                    int n_edges) {
    int wave = blockIdx.x * (blockDim.x >> 5) + (threadIdx.x >> 5);
    int lane = threadIdx.x & 31;
    if (wave >= n_edges) return;
    int s = src[wave];
    int d = dst[wave];
    const float* xr = x + (size_t)s * D_IN;
    float* ar = nsum + (size_t)d * D_IN;
#pragma unroll
    for (int i = 0; i < D_IN / 32; ++i) {
        int f = i * 32 + lane;
        unsafeAtomicAdd(&ar[f], xr[f]);
    }
    if (lane == 0) unsafeAtomicAdd(&deg[d], 1.0f);
}

// ---------------------------------------------------------------------------
// Kernel 3: fused GEMM + bias + ReLU + row L2-normalize.
// One wave computes a full 16-row x 256-col output stripe using
// V_WMMA_F32_16X16X4_F32 with 16 independent accumulator tiles.
//
// Operand layout (ISA 7.12.2, 32-bit):
//   A 16x4 : lane&15 = M, lane>>4 selects K pair -> float2 at h[m, kk]
//   B 4x16 : lane&15 = N, lane>>4 selects K pair -> {W[kk,n], W[kk+1,n]}
//   C 16x16: VGPR r holds M = r + 8*(lane>>4), N = lane&15
// ---------------------------------------------------------------------------
__global__ __launch_bounds__(128)
void sage_gemm_kernel(const float* __restrict__ x,
                      const float* __restrict__ nsum,
                      const float* __restrict__ deg,
                      const float* __restrict__ W,
                      const float* __restrict__ bias,
                      float* __restrict__ out) {
    int wave = blockIdx.x * (blockDim.x >> 5) + (threadIdx.x >> 5);
    if (wave >= N_NODES / 16) return;  // wave-uniform: EXEC stays all-1s for WMMA
    int lane = threadIdx.x & 31;
    int half = lane >> 4;
    int l16  = lane & 15;
    int m0   = wave * 16;

    const float* Arow = x    + (size_t)(m0 + l16) * D_IN;
    const float* Nrow = nsum + (size_t)(m0 + l16) * D_IN;
    float invd = 1.0f / fmaxf(deg[m0 + l16], 1.0f);

    v8f acc[16];
    const v8f vzero = {0.f, 0.f, 0.f, 0.f, 0.f, 0.f, 0.f, 0.f};
#pragma unroll
    for (int t = 0; t < 16; ++t) acc[t] = vzero;

    // K in [0,256): A = self features
    for (int k0 = 0; k0 < 256; k0 += 4) {
        int kk = k0 + 2 * half;
        v2f a = *(const v2f*)(Arow + kk);
#pragma unroll
        for (int t = 0; t < 16; ++t) {
            v2f bv;
            bv.x = W[(size_t)kk * D_OUT + t * 16 + l16];
            bv.y = W[(size_t)(kk + 1) * D_OUT + t * 16 + l16];
            acc[t] = __builtin_amdgcn_wmma_f32_16x16x4_f32(
                false, a, false, bv, (short)0, acc[t], false, false);
        }
    }
    // K in [256,512): A = neighbor mean (scale folded into the load)
    for (int k0 = 0; k0 < 256; k0 += 4) {
        int kk = k0 + 2 * half;
        v2f a = *(const v2f*)(Nrow + kk);
        a.x *= invd;
        a.y *= invd;
#pragma unroll
        for (int t = 0; t < 16; ++t) {
            v2f bv;
            bv.x = W[(size_t)(256 + kk) * D_OUT + t * 16 + l16];
            bv.y = W[(size_t)(257 + kk) * D_OUT + t * 16 + l16];
            acc[t] = __builtin_amdgcn_wmma_f32_16x16x4_f32(
                false, a, false, bv, (short)0, acc[t], false, false);
        }
    }

    float bvreg[16];
#pragma unroll
    for (int t = 0; t < 16; ++t) bvreg[t] = bias[t * 16 + l16];

    // bias + relu, per-row squared-sum partials
    float ss[8];
#pragma unroll
    for (int r = 0; r < 8; ++r) ss[r] = 0.0f;
#pragma unroll
    for (int t = 0; t < 16; ++t) {
#pragma unroll
        for (int r = 0; r < 8; ++r) {
            float v = fmaxf(acc[t][r] + bvreg[t], 0.0f);
            acc[t][r] = v;
            ss[r] += v * v;
        }
    }
    // reduce across the 16 lanes of each half-wave (xor<16 stays in-group)
#pragma unroll
    for (int r = 0; r < 8; ++r) {
        float s = ss[r];
        s += __shfl_xor(s, 1, 32);
        s += __shfl_xor(s, 2, 32);
        s += __shfl_xor(s, 4, 32);
        s += __shfl_xor(s, 8, 32);
        ss[r] = 1.0f / fmaxf(sqrtf(s), EPS_F);
    }
#pragma unroll
    for (int t = 0; t < 16; ++t) {
#pragma unroll
        for (int r = 0; r < 8; ++r) {
            int m = m0 + r + 8 * half;
            out[(size_t)m * D_OUT + t * 16 + l16] = acc[t][r] * ss[r];
        }
    }
}

// ---------------------------------------------------------------------------
extern "C" void kernel_launch(void* const* d_in, const int* in_sizes, int n_in,
                              void* d_out, int out_size, void* d_ws, size_t ws_size,
                              hipStream_t stream) {
    const float* x    = (const float*)d_in[0];
    const int*   adj  = (const int*)d_in[1];
    const float* W    = (const float*)d_in[2];
    const float* bias = (const float*)d_in[3];
    float* out = (float*)d_out;

    int n_edges = in_sizes[1] / 2;

    float* nsum = (float*)d_ws;
    float* deg  = nsum + (size_t)N_NODES * D_IN;

    long ztotal = (long)N_NODES * D_IN + N_NODES;
    sage_zero_kernel<<<2048, 256, 0, stream>>>(nsum, ztotal);

    int sblocks = (n_edges + 7) / 8;  // 8 waves (edges) per 256-thread block
    sage_scatter_kernel<<<sblocks, 256, 0, stream>>>(x, adj, adj + n_edges,
                                                     nsum, deg, n_edges);

    int gwaves  = N_NODES / 16;       // 3125 output stripes
    int gblocks = (gwaves + 3) / 4;   // 4 waves per 128-thread block
    sage_gemm_kernel<<<gblocks, 128, 0, stream>>>(x, nsum, deg, W, bias, out);
}